// RBFLayer_87668872446249
// MI455X (gfx1250) — compile-verified
//
#include <hip/hip_runtime.h>
#include <hip/hip_bf16.h>

typedef __attribute__((ext_vector_type(2))) float v2f;
typedef __attribute__((ext_vector_type(8))) float v8f;

#define GAMMA_F     0.1f
#define DIM         64
#define NCENT       1024
#define BATCH_M     65536
#define KSTEPS      16          // 64 / 4
#define NGROUPS     4           // split N=1024 into 4 groups of 16 tiles
#define NT_PER_GRP  16          // 16 n-tiles of width 16 -> 256 cols per group
#define WAVES_PER_BLOCK 8

// One wave = one (m_tile, n_group): computes a 16x256 slab of the output.
// A fragments (16x64 x-tile) are register-resident across all 16 n-tiles.
// Squared-distance via augmented-K WMMA: A_aug=[x, |x|^2, 1], B_aug=[-2c, 1, |c|^2].
__global__ __launch_bounds__(256) void rbf_wmma_f32_kernel(
    const float* __restrict__ x,
    const float* __restrict__ centers,
    float* __restrict__ out) {

  const int lane   = threadIdx.x & 31;
  const int waveib = threadIdx.x >> 5;
  const int wave_g = blockIdx.x * WAVES_PER_BLOCK + waveib;
  const int m_tile = wave_g >> 2;        // / NGROUPS
  const int n_grp  = wave_g & (NGROUPS - 1);

  const int hi  = lane >> 4;             // 0: K%4 in {0,1}, 1: K%4 in {2,3}
  const int l15 = lane & 15;
  const int row = m_tile * 16 + l15;     // A-matrix row for this lane

  // ---- Load all A fragments for the 16x64 x-tile; accumulate |x|^2 partials.
  // A 16x4 f32 layout: lanes 0-15 hold (K=0,K=1) in (v0,v1); lanes 16-31 hold (K=2,K=3).
  v2f a[KSTEPS];
  float xs_p = 0.0f;
  const float* xrow = x + row * DIM + 2 * hi;
  #pragma unroll
  for (int s = 0; s < KSTEPS; ++s) {
    v2f av = *(const v2f*)(xrow + 4 * s);
    a[s] = av;
    xs_p = fmaf(av.x, av.x, xs_p);
    xs_p = fmaf(av.y, av.y, xs_p);
  }
  // Each lane has half the K terms for its row; partner half-wave lane has the rest.
  const float xsq = xs_p + __shfl_xor(xs_p, 16, 32);

  // Augmented A step (K=64..67): [ |x|^2, 1 ] in lanes 0-15, zero-pad in lanes 16-31.
  v2f a16;
  a16.x = hi ? 0.0f : xsq;
  a16.y = hi ? 0.0f : 1.0f;

  const int n_base = n_grp * (NT_PER_GRP * 16);

  for (int t = 0; t < NT_PER_GRP; ++t) {
    const int col = n_base + t * 16 + l15;       // B-matrix column for this lane
    // B[k][n] = centers[n][k]  (B = centers^T), so fragments are contiguous loads.
    const float* crow = centers + col * DIM + 2 * hi;

    v8f acc = {};
    float cs_p = 0.0f;
    #pragma unroll
    for (int s = 0; s < KSTEPS; ++s) {
      v2f cv = *(const v2f*)(crow + 4 * s);
      cs_p = fmaf(cv.x, cv.x, cs_p);
      cs_p = fmaf(cv.y, cv.y, cs_p);
      v2f bv;
      bv.x = -2.0f * cv.x;
      bv.y = -2.0f * cv.y;
      acc = __builtin_amdgcn_wmma_f32_16x16x4_f32(
          false, a[s], false, bv, (short)0, acc, false, false);
    }
    const float csq = cs_p + __shfl_xor(cs_p, 16, 32);

    // Augmented B step: [ 1, |c|^2 ] in lanes 0-15, zero-pad in lanes 16-31.
    v2f b16;
    b16.x = hi ? 0.0f : 1.0f;
    b16.y = hi ? 0.0f : csq;
    // acc now becomes exact l2 = |x|^2 + |c|^2 - 2 x.c
    acc = __builtin_amdgcn_wmma_f32_16x16x4_f32(
        false, a16, false, b16, (short)0, acc, false, false);

    // Epilogue: clamp + exp, store. C/D layout: VGPR i, lanes 0-15 -> M=i,
    // lanes 16-31 -> M=8+i; N = lane&15.
    float* orow = out + (m_tile * 16 + 8 * hi) * NCENT + col;
    #pragma unroll
    for (int i = 0; i < 8; ++i) {
      float l2 = fmaxf(acc[i], 0.0f);
      orow[i * NCENT] = __expf(-GAMMA_F * l2);
    }
  }
}

extern "C" void kernel_launch(void* const* d_in, const int* in_sizes, int n_in,
                              void* d_out, int out_size, void* d_ws, size_t ws_size,
                              hipStream_t stream) {
  (void)in_sizes; (void)n_in; (void)out_size; (void)d_ws; (void)ws_size;
  const float* x       = (const float*)d_in[0];   // [65536, 64] f32
  const float* centers = (const float*)d_in[1];   // [1024, 64] f32
  float* out           = (float*)d_out;           // [65536, 1024] f32

  const int m_tiles     = BATCH_M / 16;                     // 4096
  const int total_waves = m_tiles * NGROUPS;                // 16384
  const int blocks      = total_waves / WAVES_PER_BLOCK;    // 2048

  rbf_wmma_f32_kernel<<<blocks, WAVES_PER_BLOCK * 32, 0, stream>>>(x, centers, out);
}